// RANSAC_89773406421385
// MI455X (gfx1250) — compile-verified
//
#include <hip/hip_runtime.h>
#include <math.h>

#define N_PTS 8192
#define N_HYP 8192

typedef __attribute__((ext_vector_type(2))) float v2f;
typedef __attribute__((ext_vector_type(8))) float v8f;

__device__ __forceinline__ unsigned hash_u32(unsigned x) {
  x ^= x >> 16; x *= 0x7feb352du;
  x ^= x >> 15; x *= 0x846ca68bu;
  x ^= x >> 16;
  return x;
}

// d1 = || R p1 + t - p2 ||^2  for rigid model (c,s,tx,ty).
// Reference sym transfer error = 2*d1; inlier test err<=2  <=>  d1<=1.
__device__ __forceinline__ float rigid_d1(float c, float s, float tx, float ty,
                                          float x1, float y1, float x2, float y2) {
  float u = c * x1 - s * y1 + tx - x2;
  float v = s * x1 + c * y1 + ty - y2;
  return u * u + v * v;
}

// ---------------------------------------------------------------------------
// Kernel 1: generate 8192 rigid hypotheses from deterministic 2-point samples
// ---------------------------------------------------------------------------
__global__ void ransac_hypgen(const float2* __restrict__ kp1,
                              const float2* __restrict__ kp2,
                              float4* __restrict__ hyp) {
  int h = blockIdx.x * blockDim.x + threadIdx.x;
  if (h >= N_HYP) return;

  unsigned base = (unsigned)h * 2654435761u;
  unsigned i0 = hash_u32(base + 1u) & (N_PTS - 1);
  unsigned i1 = hash_u32(base + 2u) & (N_PTS - 1);
  if (i1 == i0) i1 = (i1 + 1u) & (N_PTS - 1);

  float2 a1 = kp1[i0], b1 = kp1[i1];
  float2 a2 = kp2[i0], b2 = kp2[i1];

  float c1x = 0.5f * (a1.x + b1.x), c1y = 0.5f * (a1.y + b1.y);
  float c2x = 0.5f * (a2.x + b2.x), c2y = 0.5f * (a2.y + b2.y);

  float q1ax = a1.x - c1x, q1ay = a1.y - c1y;
  float q1bx = b1.x - c1x, q1by = b1.y - c1y;
  float q2ax = a2.x - c2x, q2ay = a2.y - c2y;
  float q2bx = b2.x - c2x, q2by = b2.y - c2y;

  // a = M00+M11, b = M01-M10 of M = sum q1 q2^T  (Kabsch closed form in 2D)
  float a = q1ax * q2ax + q1ay * q2ay + q1bx * q2bx + q1by * q2by;
  float b = q1ax * q2ay - q1ay * q2ax + q1bx * q2by - q1by * q2bx;
  float r = sqrtf(a * a + b * b);
  float c, s;
  if (r > 1e-20f) { c = a / r; s = b / r; } else { c = 1.0f; s = 0.0f; }
  float tx = c2x - (c * c1x - s * c1y);
  float ty = c2y - (s * c1x + c * c1y);

  hyp[h] = make_float4(c, s, tx, ty);
}

// ---------------------------------------------------------------------------
// Kernel 2: score all hypotheses vs all points with V_WMMA_F32_16X16X4_F32.
//   u-row (K=4): [c, -s, tx, -1] x [x1, y1, 1, x2]
//   v-row (K=4): [s,  c, ty, -1] x [x1, y1, 1, y2]
// One wave handles a 16-hypothesis M tile and loops over 16-point N tiles.
// ---------------------------------------------------------------------------
__global__ void ransac_score(const float2* __restrict__ kp1,
                             const float2* __restrict__ kp2,
                             const float4* __restrict__ hyp,
                             int* __restrict__ counts) {
  const int wave = threadIdx.x >> 5;           // 4 waves / block
  const int lane = threadIdx.x & 31;
  const int hyp0 = (blockIdx.x * 4 + wave) * 16;
  const int m    = lane & 15;
  const bool hi  = lane >= 16;                 // lanes 16..31 carry K=2,3

  // A operand (16x4 f32): lanes 0-15 -> K0,K1 ; lanes 16-31 -> K2,K3
  float4 hp = hyp[hyp0 + m];                   // (c, s, tx, ty)
  v2f Au, Av;
  if (!hi) { Au.x = hp.x;  Au.y = -hp.y;  Av.x = hp.y;  Av.y = hp.x;  }
  else     { Au.x = hp.z;  Au.y = -1.0f;  Av.x = hp.w;  Av.y = -1.0f; }

  float cnt[8];
#pragma unroll
  for (int j = 0; j < 8; ++j) cnt[j] = 0.0f;

  for (int n0 = 0; n0 < N_PTS; n0 += 16) {
    int p = n0 + m;
    // prefetch next point tile into cache (global_prefetch_b8)
    if (n0 + 16 < N_PTS) {
      __builtin_prefetch(&kp1[p + 16], 0, 1);
      __builtin_prefetch(&kp2[p + 16], 0, 1);
    }
    float2 p1 = kp1[p];
    float2 p2 = kp2[p];

    // B operands (4x16 f32): lanes 0-15 -> rows K0,K1 ; lanes 16-31 -> K2,K3
    v2f Bu, Bv;
    if (!hi) { Bu.x = p1.x; Bu.y = p1.y; Bv.x = p1.x; Bv.y = p1.y; }
    else     { Bu.x = 1.0f; Bu.y = p2.x; Bv.x = 1.0f; Bv.y = p2.y; }

    v8f z = {0.f, 0.f, 0.f, 0.f, 0.f, 0.f, 0.f, 0.f};
    v8f U = __builtin_amdgcn_wmma_f32_16x16x4_f32(false, Au, false, Bu,
                                                  (short)0, z, false, false);
    v8f V = __builtin_amdgcn_wmma_f32_16x16x4_f32(false, Av, false, Bv,
                                                  (short)0, z, false, false);
#pragma unroll
    for (int j = 0; j < 8; ++j) {
      float d1 = U[j] * U[j] + V[j] * V[j];
      cnt[j] += (d1 <= 1.0f) ? 1.0f : 0.0f;
    }
  }

  // D layout: lane<16 -> (M=j, N=lane); lane>=16 -> (M=j+8, N=lane-16).
  // Sum over the 16-lane N groups.
#pragma unroll
  for (int j = 0; j < 8; ++j) {
    float v = cnt[j];
    v += __shfl_xor(v, 1, 32);
    v += __shfl_xor(v, 2, 32);
    v += __shfl_xor(v, 4, 32);
    v += __shfl_xor(v, 8, 32);
    if ((lane & 15) == 0) counts[hyp0 + j + (hi ? 8 : 0)] = (int)(v + 0.5f);
  }
}

// ---------------------------------------------------------------------------
// Kernel 3: argmax over hypothesis scores (first-index tiebreak, like argmax)
//   best[] = { score, idx, c, s, tx, ty }
// ---------------------------------------------------------------------------
__global__ void ransac_best(const int* __restrict__ counts,
                            const float4* __restrict__ hyp,
                            float* __restrict__ best) {
  __shared__ int sval[256];
  __shared__ int sidx[256];
  const int t = threadIdx.x;

  int bv = -1, bi = 0;
  for (int i = t; i < N_HYP; i += 256) {
    int v = counts[i];
    if (v > bv) { bv = v; bi = i; }
  }
  sval[t] = bv; sidx[t] = bi;
  __syncthreads();
  for (int s = 128; s > 0; s >>= 1) {
    if (t < s) {
      if (sval[t + s] > sval[t] ||
          (sval[t + s] == sval[t] && sidx[t + s] < sidx[t])) {
        sval[t] = sval[t + s]; sidx[t] = sidx[t + s];
      }
    }
    __syncthreads();
  }
  if (t == 0) {
    float4 hp = hyp[sidx[0]];
    best[0] = (float)sval[0];
    best[1] = (float)sidx[0];
    best[2] = hp.x; best[3] = hp.y; best[4] = hp.z; best[5] = hp.w;
  }
}

// ---------------------------------------------------------------------------
// Kernel 4: local optimization (5 LO iters x 5 polish fits) + final output.
// Single block of 256 threads; each thread owns 32 points (strided).
// ---------------------------------------------------------------------------
__device__ __forceinline__ float block_reduce_sum(float v, float* red) {
  const int t = threadIdx.x;
  red[t] = v;
  __syncthreads();
  for (int s = 128; s > 0; s >>= 1) {
    if (t < s) red[t] += red[t + s];
    __syncthreads();
  }
  float r = red[0];
  __syncthreads();
  return r;
}

__global__ void ransac_lo(const float2* __restrict__ kp1,
                          const float2* __restrict__ kp2,
                          const float* __restrict__ best,
                          float* __restrict__ out) {
  __shared__ float red[256];
  const int t = threadIdx.x;
  const int K = N_PTS / 256;  // 32 points per thread

  float x1[32], y1[32], x2[32], y2[32];
#pragma unroll
  for (int k = 0; k < K; ++k) {
    int i = t + 256 * k;
    float2 a = kp1[i], b = kp2[i];
    x1[k] = a.x; y1[k] = a.y; x2[k] = b.x; y2[k] = b.y;
  }

  // Unweighted centroids (reference uses plain means regardless of weights)
  float sx1 = 0.f, sy1 = 0.f, sx2 = 0.f, sy2 = 0.f;
#pragma unroll
  for (int k = 0; k < K; ++k) { sx1 += x1[k]; sy1 += y1[k]; sx2 += x2[k]; sy2 += y2[k]; }
  const float inv_n = 1.0f / (float)N_PTS;
  float c1x = block_reduce_sum(sx1, red) * inv_n;
  float c1y = block_reduce_sum(sy1, red) * inv_n;
  float c2x = block_reduce_sum(sx2, red) * inv_n;
  float c2y = block_reduce_sum(sy2, red) * inv_n;

  float mc = best[2], ms = best[3], mtx = best[4], mty = best[5];
  float score = best[0];

  for (int lo = 0; lo < 5; ++lo) {
    // weights = inliers of current best model
    float w[32];
#pragma unroll
    for (int k = 0; k < K; ++k) {
      float d1 = rigid_d1(mc, ms, mtx, mty, x1[k], y1[k], x2[k], y2[k]);
      w[k] = (d1 <= 1.0f) ? 1.0f : 0.0f;
    }

    float cc = 1.f, cs = 0.f, ctx = 0.f, cty = 0.f;
    for (int it = 0; it < 5; ++it) {
      if (it > 0) {
        // soft weights from non-squared sym error: e = 2*sqrt(d1),
        // w = exp(-e / (2*SOFT_INL_TH^2)) = exp(-e/18)
#pragma unroll
        for (int k = 0; k < K; ++k) {
          float d1 = rigid_d1(cc, cs, ctx, cty, x1[k], y1[k], x2[k], y2[k]);
          float e = 2.0f * sqrtf(fmaxf(d1, 1e-12f));
          w[k] = expf(-e * (1.0f / 18.0f));
        }
      }
      // weighted Kabsch (weights applied to q2 only, per reference)
      float ap = 0.f, bp = 0.f;
#pragma unroll
      for (int k = 0; k < K; ++k) {
        float q1x = x1[k] - c1x, q1y = y1[k] - c1y;
        float q2x = (x2[k] - c2x) * w[k], q2y = (y2[k] - c2y) * w[k];
        ap += q1x * q2x + q1y * q2y;
        bp += q1x * q2y - q1y * q2x;
      }
      float A = block_reduce_sum(ap, red);
      float B = block_reduce_sum(bp, red);
      float r = sqrtf(A * A + B * B);
      if (r > 1e-20f) { cc = A / r; cs = B / r; } else { cc = 1.0f; cs = 0.0f; }
      ctx = c2x - (cc * c1x - cs * c1y);
      cty = c2y - (cs * c1x + cc * c1y);
    }

    // score candidate
    float cp = 0.f;
#pragma unroll
    for (int k = 0; k < K; ++k) {
      float d1 = rigid_d1(cc, cs, ctx, cty, x1[k], y1[k], x2[k], y2[k]);
      cp += (d1 <= 1.0f) ? 1.0f : 0.0f;
    }
    float s_lo = block_reduce_sum(cp, red);
    if (s_lo > score) { mc = cc; ms = cs; mtx = ctx; mty = cty; score = s_lo; }
  }

  // Emit 3x3 model then inlier mask (as 0/1 floats)
  if (t == 0) {
    out[0] = mc;  out[1] = -ms; out[2] = mtx;
    out[3] = ms;  out[4] = mc;  out[5] = mty;
    out[6] = 0.f; out[7] = 0.f; out[8] = 1.f;
  }
#pragma unroll
  for (int k = 0; k < K; ++k) {
    int i = t + 256 * k;
    float d1 = rigid_d1(mc, ms, mtx, mty, x1[k], y1[k], x2[k], y2[k]);
    out[9 + i] = (d1 <= 1.0f) ? 1.0f : 0.0f;
  }
}

// ---------------------------------------------------------------------------
extern "C" void kernel_launch(void* const* d_in, const int* in_sizes, int n_in,
                              void* d_out, int out_size, void* d_ws, size_t ws_size,
                              hipStream_t stream) {
  (void)in_sizes; (void)n_in; (void)out_size; (void)ws_size;
  const float2* kp1 = (const float2*)d_in[0];
  const float2* kp2 = (const float2*)d_in[1];
  float* out = (float*)d_out;

  char* ws = (char*)d_ws;
  float4* hyp   = (float4*)(ws);                                   // 128 KB
  int*    cnts  = (int*)(ws + (size_t)N_HYP * sizeof(float4));     // 32 KB
  float*  best  = (float*)(ws + (size_t)N_HYP * sizeof(float4)
                              + (size_t)N_HYP * sizeof(int));      // 6 floats

  hipLaunchKernelGGL(ransac_hypgen, dim3(N_HYP / 256), dim3(256), 0, stream,
                     kp1, kp2, hyp);
  hipLaunchKernelGGL(ransac_score, dim3(N_HYP / 64), dim3(128), 0, stream,
                     kp1, kp2, hyp, cnts);
  hipLaunchKernelGGL(ransac_best, dim3(1), dim3(256), 0, stream,
                     cnts, hyp, best);
  hipLaunchKernelGGL(ransac_lo, dim3(1), dim3(256), 0, stream,
                     kp1, kp2, best, out);
}